// GraphSAGEClassifier_31138512896066
// MI455X (gfx1250) — compile-verified
//
#include <hip/hip_runtime.h>
#include <hip/hip_bf16.h>

#define N_NODES 65536
#define N_EDGES 1048576
#define N_GRAPHS 64
#define DIM 64
#define OUTD 8

typedef float v2f __attribute__((ext_vector_type(2)));
typedef float v8f __attribute__((ext_vector_type(8)));

// ---------------- utility kernels ----------------

__global__ void zero_kernel(float* p, int n) {
    int i = blockIdx.x * blockDim.x + threadIdx.x;
    if (i < n) p[i] = 0.0f;
}

__global__ void degree_kernel(const int* __restrict__ dst, float* __restrict__ deg) {
    int e = blockIdx.x * blockDim.x + threadIdx.x;
    if (e < N_EDGES) atomicAdd(&deg[dst[e]], 1.0f);
}

__global__ void invdeg_kernel(float* deg) {
    int n = blockIdx.x * blockDim.x + threadIdx.x;
    if (n < N_NODES) {
        float d = deg[n];
        deg[n] = 1.0f / fmaxf(d, 1.0f);
    }
}

// One edge handled by 16 consecutive threads, each moving 4 floats (float4).
__global__ void scatter_kernel(const float* __restrict__ h,
                               const int* __restrict__ src,
                               const int* __restrict__ dst,
                               float* __restrict__ sum) {
    int tid = blockIdx.x * blockDim.x + threadIdx.x;   // 16M threads total
    int e = tid >> 4;
    int dbase = (tid & 15) * 4;
    if (e < N_EDGES) {
        int s = src[e];
        int d = dst[e];
        const float4 v = *(const float4*)(h + (size_t)s * DIM + dbase);
        float* o = sum + (size_t)d * DIM + dbase;
        atomicAdd(o + 0, v.x);
        atomicAdd(o + 1, v.y);
        atomicAdd(o + 2, v.z);
        atomicAdd(o + 3, v.w);
    }
}

// ---------------- fused SAGE layer: Out = relu(H@Wself + (Agg*invdeg)@Wneigh + b) ----------------
// Block = 256 threads = 8 wave32s; each wave computes a 16x64 output tile with
// V_WMMA_F32_16X16X4_F32 (fp32-exact): 16 k-steps x 4 col-tiles x 2 (self+neigh).
// Weights live in LDS k-pair interleaved so each B fragment is one ds_load_b64:
//   wpk[(kp*64 + c)*2 + j] = W[2*kp + j][c]   (kp = k/2, j in {0,1})

__global__ __launch_bounds__(256) void sage_layer_kernel(
    const float* __restrict__ H, const float* __restrict__ Agg,
    const float* __restrict__ invdeg,
    const float* __restrict__ Wself, const float* __restrict__ Wneigh,
    const float* __restrict__ bias, float* __restrict__ Out)
{
    __shared__ float ws[DIM * DIM];   // k-pair interleaved W_self
    __shared__ float wn[DIM * DIM];   // k-pair interleaved W_neigh
    __shared__ float bs[DIM];

    int tid = threadIdx.x;
    // stage + repack: i enumerates (kp, c)
    for (int i = tid; i < (DIM / 2) * DIM; i += 256) {
        int kp = i >> 6;          // 0..31
        int c  = i & 63;
        ws[i * 2 + 0] = Wself[(kp * 2 + 0) * DIM + c];
        ws[i * 2 + 1] = Wself[(kp * 2 + 1) * DIM + c];
        wn[i * 2 + 0] = Wneigh[(kp * 2 + 0) * DIM + c];
        wn[i * 2 + 1] = Wneigh[(kp * 2 + 1) * DIM + c];
    }
    if (tid < DIM) bs[tid] = bias[tid];
    __syncthreads();

    int wave = tid >> 5;
    int lane = tid & 31;
    int rowbase = (blockIdx.x * 8 + wave) * 16;

    // A-fragment layout (32-bit A 16x4): lane&15 = M, lane>>4 selects K-half.
    int r = rowbase + (lane & 15);
    int khalf = lane >> 4;          // 0 or 1
    int col0 = lane & 15;           // B/C: lane&15 = N within tile

    float iv = invdeg[r];

    const float* hrow = H + (size_t)r * DIM;
    const float* arow = Agg + (size_t)r * DIM;

    v8f acc[4];
#pragma unroll
    for (int nt = 0; nt < 4; ++nt)
#pragma unroll
        for (int i = 0; i < 8; ++i)
            acc[nt][i] = 0.0f;

#pragma unroll
    for (int ks = 0; ks < 16; ++ks) {
        int k0 = ks * 4 + khalf * 2;      // even
        int kp = ks * 2 + khalf;          // k-pair index
        v2f a  = *(const v2f*)(hrow + k0);        // {K=k0, K=k0+1} for row M=r
        v2f an = *(const v2f*)(arow + k0);
        an.x *= iv;
        an.y *= iv;
#pragma unroll
        for (int nt = 0; nt < 4; ++nt) {
            int c = nt * 16 + col0;
            v2f b  = *(const v2f*)&ws[(kp * DIM + c) * 2];   // one ds_load_b64
            v2f bn = *(const v2f*)&wn[(kp * DIM + c) * 2];
            acc[nt] = __builtin_amdgcn_wmma_f32_16x16x4_f32(
                false, a, false, b, (short)0, acc[nt], false, false);
            acc[nt] = __builtin_amdgcn_wmma_f32_16x16x4_f32(
                false, an, false, bn, (short)0, acc[nt], false, false);
        }
    }

    // Epilogue: C/D layout — VGPR i: lanes 0-15 -> M=i, lanes 16-31 -> M=8+i; N = lane&15.
    int nhalf = lane >> 4;
#pragma unroll
    for (int nt = 0; nt < 4; ++nt) {
        int c = nt * 16 + col0;
        float bv = bs[c];
#pragma unroll
        for (int i = 0; i < 8; ++i) {
            int row = rowbase + i + nhalf * 8;
            float v = acc[nt][i] + bv;
            Out[(size_t)row * DIM + c] = fmaxf(v, 0.0f);
        }
    }
}

// ---------------- pooling: segment-sum over sorted graph_ids ----------------
// 64 threads per block (one per feature dim), 256 nodes per block; running
// accumulator flushed on graph-id change (ids are sorted -> few atomics).

__global__ void pool_kernel(const float* __restrict__ h2,
                            const int* __restrict__ gid,
                            float* __restrict__ hg) {
    int n0 = blockIdx.x * 256;
    int d = threadIdx.x;   // 0..63
    float acc = 0.0f;
    int cur = gid[n0];
    for (int i = 0; i < 256; ++i) {
        int n = n0 + i;
        int g = gid[n];
        if (g != cur) {
            atomicAdd(&hg[cur * DIM + d], acc);
            acc = 0.0f;
            cur = g;
        }
        acc += h2[(size_t)n * DIM + d];
    }
    atomicAdd(&hg[cur * DIM + d], acc);
}

// ---------------- head: out = hg @ Wlin + blin ----------------

__global__ void final_kernel(const float* __restrict__ hg,
                             const float* __restrict__ Wlin,
                             const float* __restrict__ blin,
                             float* __restrict__ out) {
    int t = threadIdx.x;      // 512 threads
    int i = t >> 3;           // graph 0..63
    int j = t & 7;            // out dim 0..7
    float acc = blin[j];
    for (int k = 0; k < DIM; ++k)
        acc += hg[i * DIM + k] * Wlin[k * OUTD + j];
    out[i * OUTD + j] = acc;
}

// ---------------- launch ----------------

extern "C" void kernel_launch(void* const* d_in, const int* in_sizes, int n_in,
                              void* d_out, int out_size, void* d_ws, size_t ws_size,
                              hipStream_t stream) {
    const float* feat = (const float*)d_in[0];
    const int*   src  = (const int*)d_in[1];
    const int*   dst  = (const int*)d_in[2];
    const int*   gid  = (const int*)d_in[3];
    const float* W1s  = (const float*)d_in[4];
    const float* W1n  = (const float*)d_in[5];
    const float* b1   = (const float*)d_in[6];
    const float* W2s  = (const float*)d_in[7];
    const float* W2n  = (const float*)d_in[8];
    const float* b2   = (const float*)d_in[9];
    const float* Wlin = (const float*)d_in[10];
    const float* blin = (const float*)d_in[11];
    float* out = (float*)d_out;

    float* deg  = (float*)d_ws;                         // N floats (becomes invdeg)
    float* bufA = deg + N_NODES;                        // N*64 (agg buffer)
    float* bufB = bufA + (size_t)N_NODES * DIM;         // N*64 (h1)
    float* bufC = bufB + (size_t)N_NODES * DIM;         // N*64 (h2)
    float* hg   = bufC + (size_t)N_NODES * DIM;         // 64*64

    const int NF = N_NODES * DIM;

    // degree + invdeg
    zero_kernel<<<(N_NODES + 255) / 256, 256, 0, stream>>>(deg, N_NODES);
    degree_kernel<<<(N_EDGES + 255) / 256, 256, 0, stream>>>(dst, deg);
    invdeg_kernel<<<(N_NODES + 255) / 256, 256, 0, stream>>>(deg);

    // layer 1 aggregation: bufA = segment_sum(feat[src], dst)
    zero_kernel<<<(NF + 255) / 256, 256, 0, stream>>>(bufA, NF);
    scatter_kernel<<<(N_EDGES * 16) / 256, 256, 0, stream>>>(feat, src, dst, bufA);

    // layer 1: bufB = relu(feat@W1s + (bufA*invdeg)@W1n + b1)
    sage_layer_kernel<<<N_NODES / 128, 256, 0, stream>>>(feat, bufA, deg, W1s, W1n, b1, bufB);

    // layer 2 aggregation: bufA = segment_sum(bufB[src], dst)
    zero_kernel<<<(NF + 255) / 256, 256, 0, stream>>>(bufA, NF);
    scatter_kernel<<<(N_EDGES * 16) / 256, 256, 0, stream>>>(bufB, src, dst, bufA);

    // layer 2: bufC = relu(bufB@W2s + (bufA*invdeg)@W2n + b2)
    sage_layer_kernel<<<N_NODES / 128, 256, 0, stream>>>(bufB, bufA, deg, W2s, W2n, b2, bufC);

    // pooling + head
    zero_kernel<<<(N_GRAPHS * DIM + 255) / 256, 256, 0, stream>>>(hg, N_GRAPHS * DIM);
    pool_kernel<<<N_NODES / 256, 64, 0, stream>>>(bufC, gid, hg);
    final_kernel<<<1, N_GRAPHS * OUTD, 0, stream>>>(hg, Wlin, blin, out);
}